// SupConLoss_63513976373818
// MI455X (gfx1250) — compile-verified
//
#include <hip/hip_runtime.h>
#include <hip/hip_bf16.h>

// SupCon-style audio/visual contrastive loss, fused flash-LSE on gfx1250.
//
//  S_v = Vv @ Av^T, S_a = Ai @ Vi^T   (each 8192x8192, K=64, never materialized)
//  out = mean(lse(S_v) - diag(S_v)) + mean(lse(S_a) - diag(S_a))
//
// Matrix core: v_wmma_f32_16x16x4_f32 (exact fp32), 2 independent 8-step
// accumulator chains per 16x16 tile. Key chunks staged into LDS with
// global_load_async_to_lds_b128 (ASYNCcnt) — CDNA5 async copy path.

typedef float v2f __attribute__((ext_vector_type(2)));
typedef float v8f __attribute__((ext_vector_type(8)));

#define NROWS 8192
#define DFULL 128
#define DSEL  64
#define CHUNK 128   // key rows staged in LDS per iteration (32 KB)

// ---------------------------------------------------------------- gather ----
__global__ void supcon_gather_kernel(const float* __restrict__ a,
                                     const float* __restrict__ v,
                                     const int* __restrict__ aidx,
                                     const int* __restrict__ iidx,
                                     float* __restrict__ G, int total)
{
    int tid = blockIdx.x * blockDim.x + threadIdx.x;
    if (tid >= total) return;
    const int MAT = NROWS * DSEL;
    int mtx = tid / MAT;            // 0:Av 1:Vv 2:Vi 3:Ai
    int rem = tid - mtx * MAT;
    int r = rem >> 6;               // /DSEL
    int c = rem & (DSEL - 1);
    const float* src = (mtx == 1 || mtx == 2) ? v : a;
    const int*   idx = (mtx < 2) ? aidx : iidx;
    G[tid] = src[r * DFULL + idx[c]];
}

// ------------------------------------------------------- fused WMMA + LSE ---
// block = 128 threads = 4 waves; each wave owns a 16-row strip of Q.
// gridDim.x = 256: blockIdx.x&1 selects which of the two score matrices,
// blockIdx.x>>1 selects the 64-row strip.
__global__ __launch_bounds__(128)
void supcon_lse_kernel(const float* __restrict__ Q0, const float* __restrict__ K0,
                       const float* __restrict__ Q1, const float* __restrict__ K1,
                       float* __restrict__ partials)
{
    const int mat    = blockIdx.x & 1;
    const int strip  = blockIdx.x >> 1;
    const float* Q   = mat ? Q1 : Q0;
    const float* K   = mat ? K1 : K0;
    const int wave   = threadIdx.x >> 5;
    const int lane   = threadIdx.x & 31;
    const int lane16 = lane & 15;
    const int koff   = (lane < 16) ? 0 : 2;        // K-slot pair per half-wave
    const int rm     = strip * 64 + wave * 16;     // first row of this wave

    __shared__ float sK[CHUNK * DSEL];             // 32 KB staged key chunk

    // ---- preload A-operand (Q) fragments for all 16 K-steps: 32 VGPRs ----
    // A 16x4 f32 layout: lane L holds M=L%16; VGPR0/1 = K = koff+0 / koff+1.
    const float* qrow = Q + (size_t)(rm + lane16) * DSEL;
    v2f qf[16];
#pragma unroll
    for (int s = 0; s < 16; ++s)
        qf[s] = *(const v2f*)(qrow + s * 4 + koff);

    // ---- diagonal element for row (rm + lane16), duplicated in half-waves --
    const float* krow = K + (size_t)(rm + lane16) * DSEL;
    float dval = 0.f;
#pragma unroll 8
    for (int j = 0; j < DSEL; ++j)
        dval = fmaf(qrow[j], krow[j], dval);

    // ---- per-lane online logsumexp state: acc[v] covers row rm+v(+8) ------
    float mrun[8], srun[8];
#pragma unroll
    for (int v = 0; v < 8; ++v) { mrun[v] = -1e30f; srun[v] = 0.f; }

    // LDS byte address of sK (flat->LDS: low 32 bits are the LDS offset)
    const unsigned int lbase = (unsigned int)(uintptr_t)(void*)sK;

    for (int c0 = 0; c0 < NROWS; c0 += CHUNK) {
        // async stage: 128 rows x 64 f32 -> LDS, 16B per lane per op,
        // 16 uniform iterations per thread (EXEC all-ones throughout).
        const unsigned long long gbase =
            (unsigned long long)(uintptr_t)(K + (size_t)c0 * DSEL);
        for (int i = threadIdx.x; i < (CHUNK * DSEL) / 4; i += 128) {
            unsigned int       laddr = lbase + (unsigned int)i * 16u;
            unsigned long long gaddr = gbase + (unsigned long long)i * 16ull;
            asm volatile("global_load_async_to_lds_b128 %0, %1, off"
                         :: "v"(laddr), "v"(gaddr) : "memory");
        }
        asm volatile("s_wait_asynccnt 0x0" ::: "memory");
        __syncthreads();

        // prefetch next chunk (one 256B slice per thread -> global_prefetch)
        if (c0 + CHUNK < NROWS) {
            const char* np = (const char*)(K + (size_t)(c0 + CHUNK) * DSEL);
            __builtin_prefetch(np + (size_t)threadIdx.x * 256, 0, 1);
        }

        for (int t = 0; t < CHUNK / 16; ++t) {
            // B 4x16 f32 layout: lane L holds N=L%16; VGPR0/1 = K = koff+0/1.
            const float* kbase = sK + (t * 16 + lane16) * DSEL + koff;
            v8f acc0 = {0.f, 0.f, 0.f, 0.f, 0.f, 0.f, 0.f, 0.f};
            v8f acc1 = {0.f, 0.f, 0.f, 0.f, 0.f, 0.f, 0.f, 0.f};
#pragma unroll
            for (int s = 0; s < 8; ++s) {
                v2f bf0 = *(const v2f*)(kbase + s * 4);
                v2f bf1 = *(const v2f*)(kbase + (s + 8) * 4);
                acc0 = __builtin_amdgcn_wmma_f32_16x16x4_f32(
                           false, qf[s], false, bf0, (short)0, acc0, false, false);
                acc1 = __builtin_amdgcn_wmma_f32_16x16x4_f32(
                           false, qf[s + 8], false, bf1, (short)0, acc1, false, false);
            }
            // online (max, sumexp) update — purely per-lane
#pragma unroll
            for (int v = 0; v < 8; ++v) {
                float val = acc0[v] + acc1[v];
                float nm  = fmaxf(mrun[v], val);
                srun[v]   = srun[v] * __expf(mrun[v] - nm) + __expf(val - nm);
                mrun[v]   = nm;
            }
        }
        __syncthreads();
    }

    // ---- merge the 16 column-lanes of each half-wave (rows rm..+7 / +8..15)
#pragma unroll
    for (int v = 0; v < 8; ++v) {
#pragma unroll
        for (int msk = 1; msk <= 8; msk <<= 1) {
            float om = __shfl_xor(mrun[v], msk, 32);
            float os = __shfl_xor(srun[v], msk, 32);
            float nm = fmaxf(mrun[v], om);
            srun[v]  = srun[v] * __expf(mrun[v] - nm) + os * __expf(om - nm);
            mrun[v]  = nm;
        }
    }

    // lse sum for this half-wave's 8 rows, then add the other half's
    float lsum = 0.f;
#pragma unroll
    for (int v = 0; v < 8; ++v)
        lsum += mrun[v] + __logf(srun[v]);
    lsum += __shfl_xor(lsum, 16, 32);              // all 16 rows

    // diagonal sum over the wave's 16 rows (lanes 0..15 hold distinct rows)
    float dsum = dval;
#pragma unroll
    for (int msk = 1; msk <= 8; msk <<= 1)
        dsum += __shfl_xor(dsum, msk, 32);

    if (lane == 0)
        partials[blockIdx.x * 4 + wave] = lsum - dsum;
}

// ------------------------------------------------- deterministic finalize ---
__global__ void supcon_finalize_kernel(const float* __restrict__ partials,
                                       float* __restrict__ out)
{
    __shared__ float red[256];
    float s = 0.f;
    for (int i = threadIdx.x; i < 1024; i += 256)
        s += partials[i];
    red[threadIdx.x] = s;
    __syncthreads();
    for (int w = 128; w > 0; w >>= 1) {
        if (threadIdx.x < w) red[threadIdx.x] += red[threadIdx.x + w];
        __syncthreads();
    }
    if (threadIdx.x == 0) out[0] = red[0] / (float)NROWS;
}

// ---------------------------------------------------------------- launch ----
extern "C" void kernel_launch(void* const* d_in, const int* in_sizes, int n_in,
                              void* d_out, int out_size, void* d_ws, size_t ws_size,
                              hipStream_t stream)
{
    const float* a    = (const float*)d_in[0];   // (8192,128) f32
    const float* v    = (const float*)d_in[1];   // (8192,128) f32
    const int*   aidx = (const int*)d_in[2];     // (64,)
    const int*   iidx = (const int*)d_in[3];     // (64,)
    // d_in[4] = labels, unused by the loss

    float* W        = (float*)d_ws;
    float* partials = W;                          // 1024 floats
    float* G        = W + 1024;                   // 4 x 8192 x 64 floats (8 MB)
    const size_t MAT = (size_t)NROWS * DSEL;

    const int total = 4 * NROWS * DSEL;
    supcon_gather_kernel<<<(total + 255) / 256, 256, 0, stream>>>(
        a, v, aidx, iidx, G, total);

    // S_v: Q = Vv (G1), K = Av (G0);   S_a: Q = Ai (G3), K = Vi (G2)
    supcon_lse_kernel<<<256, 128, 0, stream>>>(
        G + 1 * MAT, G + 0 * MAT,
        G + 3 * MAT, G + 2 * MAT,
        partials);

    supcon_finalize_kernel<<<1, 256, 0, stream>>>(partials, (float*)d_out);
}